// GAT_73426760892928
// MI455X (gfx1250) — compile-verified
//
#include <hip/hip_runtime.h>
#include <math.h>

#define NN 20000
#define EE 320000
#define ETOT (EE + NN)
#define BB 64
#define FIN 128
#define HH 512

typedef float v2f __attribute__((ext_vector_type(2)));
typedef float v8f __attribute__((ext_vector_type(8)));

// ---------------- atomic helper (HW float atomic, global_atomic_add_f32) ----
__device__ __forceinline__ void atomicAddF(float* p, float v) {
  unsafeAtomicAdd(p, v);
}

// ---------------- WMMA f32 GEMM: Y[M,512] = X[M,K] @ W[K,512] ----------------
// One wave computes a 16x64 tile via 4 accumulators of V_WMMA_F32_16X16X4_F32.
// 8 waves/block -> a full 16x512 row stripe. grid.x = M/16 (M=20000 -> 1250).
__global__ void gemm_wmma_k(const float* __restrict__ X, const float* __restrict__ W,
                            float* __restrict__ Y, int K) {
  const int lane = threadIdx.x & 31;
  const int wave = threadIdx.x >> 5;     // 0..7
  const int row0 = blockIdx.x << 4;      // 16-row tile
  const int col0 = wave << 6;            // 64 cols per wave
  const int m    = lane & 15;
  const int koff = (lane >> 4) << 1;     // lanes 16..31 hold K+2,K+3

  v8f acc0 = {0.f,0.f,0.f,0.f,0.f,0.f,0.f,0.f};
  v8f acc1 = acc0, acc2 = acc0, acc3 = acc0;

  const float* xrow = X + (size_t)(row0 + m) * K;
  for (int k0 = 0; k0 < K; k0 += 4) {
    const int ka = k0 + koff;
    v2f a; a.x = xrow[ka]; a.y = xrow[ka + 1];          // A: 16x4, lane->M, vgpr->K
    const float* wp = W + (size_t)ka * HH + col0 + m;   // B: 4x16, lane->N, vgpr->K
    v2f b0; b0.x = wp[0];   b0.y = wp[HH];
    v2f b1; b1.x = wp[16];  b1.y = wp[HH + 16];
    v2f b2; b2.x = wp[32];  b2.y = wp[HH + 32];
    v2f b3; b3.x = wp[48];  b3.y = wp[HH + 48];
    acc0 = __builtin_amdgcn_wmma_f32_16x16x4_f32(false, a, false, b0, (short)0, acc0, false, false);
    acc1 = __builtin_amdgcn_wmma_f32_16x16x4_f32(false, a, false, b1, (short)0, acc1, false, false);
    acc2 = __builtin_amdgcn_wmma_f32_16x16x4_f32(false, a, false, b2, (short)0, acc2, false, false);
    acc3 = __builtin_amdgcn_wmma_f32_16x16x4_f32(false, a, false, b3, (short)0, acc3, false, false);
  }

  // C/D layout: VGPR j -> M = (lane>=16 ? 8 : 0) + j ; N = col0 + (lane&15)
  const int r0 = row0 + ((lane >> 4) << 3);
  const int cb = col0 + m;
  float* y0 = Y + (size_t)r0 * HH + cb;
#pragma unroll
  for (int j = 0; j < 8; ++j) {
    y0[(size_t)j * HH +  0] = acc0[j];
    y0[(size_t)j * HH + 16] = acc1[j];
    y0[(size_t)j * HH + 32] = acc2[j];
    y0[(size_t)j * HH + 48] = acc3[j];
  }
}

// ---------------- per-node attention scores: e_src = h.a_s, e_dst = h.a_d ----
__global__ void node_scores_k(const float* __restrict__ h,
                              const float* __restrict__ as_, const float* __restrict__ ad_,
                              float* __restrict__ es, float* __restrict__ ed) {
  const int lane = threadIdx.x & 31;
  const int wave = threadIdx.x >> 5;
  const int node = blockIdx.x * 8 + wave;
  if (node >= NN) return;
  const float* hp = h + (size_t)node * HH;
  float s = 0.f, d = 0.f;
  for (int c = lane; c < HH; c += 32) {
    float v = hp[c];
    s += v * as_[c];
    d += v * ad_[c];
  }
#pragma unroll
  for (int off = 16; off > 0; off >>= 1) {
    s += __shfl_xor(s, off, 32);
    d += __shfl_xor(d, off, 32);
  }
  if (lane == 0) { es[node] = s; ed[node] = d; }
}

// ---------------- fills ----------------
__global__ void fill_f32_k(float* p, float v, int n) {
  int i = blockIdx.x * blockDim.x + threadIdx.x;
  if (i < n) p[i] = v;
}
__global__ void fill_i32_k(int* p, int v, int n) {
  int i = blockIdx.x * blockDim.x + threadIdx.x;
  if (i < n) p[i] = v;
}

// ---------------- CSR build (incoming edges per dst) ----------------
__device__ __forceinline__ void edge_sd(const int* __restrict__ ei, int i, int& s, int& d) {
  if (i < EE) { s = ei[i]; d = ei[EE + i]; }
  else        { s = i - EE; d = i - EE; }          // self loops
}

__global__ void deg_k(const int* __restrict__ ei, int* __restrict__ deg) {
  int i = blockIdx.x * blockDim.x + threadIdx.x;
  if (i >= ETOT) return;
  int s, d; edge_sd(ei, i, s, d);
  atomicAdd(&deg[d], 1);
}

#define SCAN_T 256
__global__ void scan_rowptr_k(const int* __restrict__ deg, int* __restrict__ rowptr) {
  // single block exclusive scan over NN degrees -> rowptr[NN+1]
  __shared__ int ssum[SCAN_T];
  const int t  = threadIdx.x;
  const int CH = (NN + SCAN_T - 1) / SCAN_T;   // 79
  const int base = t * CH;
  int s = 0;
  for (int j = 0; j < CH; ++j) {
    int idx = base + j;
    if (idx < NN) s += deg[idx];
  }
  ssum[t] = s;
  __syncthreads();
  for (int off = 1; off < SCAN_T; off <<= 1) {   // Hillis-Steele inclusive scan
    int v = (t >= off) ? ssum[t - off] : 0;
    __syncthreads();
    ssum[t] += v;
    __syncthreads();
  }
  int run = (t == 0) ? 0 : ssum[t - 1];
  for (int j = 0; j < CH; ++j) {
    int idx = base + j;
    if (idx < NN) { rowptr[idx] = run; run += deg[idx]; }
  }
  if (t == SCAN_T - 1) rowptr[NN] = run;
}

__global__ void cursor_init_k(const int* __restrict__ rowptr, int* __restrict__ cursor) {
  int i = blockIdx.x * blockDim.x + threadIdx.x;
  if (i < NN) cursor[i] = rowptr[i];
}

__global__ void scatter_k(const int* __restrict__ ei, int* __restrict__ cursor,
                          int* __restrict__ colsrc) {
  int i = blockIdx.x * blockDim.x + threadIdx.x;
  if (i >= ETOT) return;
  int s, d; edge_sd(ei, i, s, d);
  int pos = atomicAdd(&cursor[d], 1);
  colsrc[pos] = s;
}

// ------- fused per-node GAT: softmax over incoming edges + gather-aggregate --
// one wave per destination node; no atomics; coalesced float4 row gathers.
__global__ void gat_node_k(const int* __restrict__ rowptr, const int* __restrict__ colsrc,
                           const float* __restrict__ es, const float* __restrict__ ed,
                           const float* __restrict__ h, const float* __restrict__ bias,
                           float* __restrict__ out) {
  const int lane = threadIdx.x & 31;
  const int wave = threadIdx.x >> 5;
  const int d = blockIdx.x * 8 + wave;
  if (d >= NN) return;
  const int beg = rowptr[d], end = rowptr[d + 1];
  const float edv = ed[d];

  // pass 1: running max of leaky_relu(e_src + e_dst)
  float mx = -3.0e38f;
  for (int j = beg + lane; j < end; j += 32) {
    float e = es[colsrc[j]] + edv;
    e = (e > 0.f) ? e : 0.2f * e;
    mx = fmaxf(mx, e);
  }
#pragma unroll
  for (int off = 16; off > 0; off >>= 1) mx = fmaxf(mx, __shfl_xor(mx, off, 32));

  // pass 2: sum of exp
  float z = 0.f;
  for (int j = beg + lane; j < end; j += 32) {
    float e = es[colsrc[j]] + edv;
    e = (e > 0.f) ? e : 0.2f * e;
    z += expf(e - mx);
  }
#pragma unroll
  for (int off = 16; off > 0; off >>= 1) z += __shfl_xor(z, off, 32);
  const float invz = 1.0f / (z + 1e-16f);

  // pass 3: out[d] = sum_j alpha_j * h[src_j]  (lane owns 16 channels)
  float4 a0 = {0.f,0.f,0.f,0.f}, a1 = a0, a2 = a0, a3 = a0;
  for (int j = beg; j < end; ++j) {
    const int s = colsrc[j];
    const int jp = j + 4;                            // software prefetch ahead
    if (jp < end) {
      const float* pf = h + (size_t)colsrc[jp] * HH + lane * 16;
      __builtin_prefetch(pf, 0, 3);                  // -> global_prefetch_b8
    }
    float e = es[s] + edv;
    e = (e > 0.f) ? e : 0.2f * e;
    const float w = expf(e - mx) * invz;
    const float4* hs = reinterpret_cast<const float4*>(h + (size_t)s * HH);
    float4 v0 = hs[lane], v1 = hs[lane + 32], v2 = hs[lane + 64], v3 = hs[lane + 96];
    a0.x += w * v0.x; a0.y += w * v0.y; a0.z += w * v0.z; a0.w += w * v0.w;
    a1.x += w * v1.x; a1.y += w * v1.y; a1.z += w * v1.z; a1.w += w * v1.w;
    a2.x += w * v2.x; a2.y += w * v2.y; a2.z += w * v2.z; a2.w += w * v2.w;
    a3.x += w * v3.x; a3.y += w * v3.y; a3.z += w * v3.z; a3.w += w * v3.w;
  }
  const float4* bp = reinterpret_cast<const float4*>(bias);
  float4 b0 = bp[lane], b1 = bp[lane + 32], b2 = bp[lane + 64], b3 = bp[lane + 96];
  a0.x += b0.x; a0.y += b0.y; a0.z += b0.z; a0.w += b0.w;
  a1.x += b1.x; a1.y += b1.y; a1.z += b1.z; a1.w += b1.w;
  a2.x += b2.x; a2.y += b2.y; a2.z += b2.z; a2.w += b2.w;
  a3.x += b3.x; a3.y += b3.y; a3.z += b3.z; a3.w += b3.w;
  float4* o4 = reinterpret_cast<float4*>(out + (size_t)d * HH);
  o4[lane] = a0; o4[lane + 32] = a1; o4[lane + 64] = a2; o4[lane + 96] = a3;
}

// ---------------- batchnorm stats: column sums over N rows ----------------
#define BN_ROWS 40   // 500 blocks * 40 rows = 20000
__global__ void bn_stats_k(const float* __restrict__ a,
                           float* __restrict__ colsum, float* __restrict__ colsq) {
  const int c  = threadIdx.x;            // blockDim = 512
  const int r0 = blockIdx.x * BN_ROWS;
  float s = 0.f, sq = 0.f;
  for (int r = 0; r < BN_ROWS; ++r) {
    float v = a[(size_t)(r0 + r) * HH + c];
    s += v; sq += v * v;
  }
  atomicAddF(&colsum[c], s);
  atomicAddF(&colsq[c], sq);
}

// ---------------- normalize + relu (in place) + mean-pool accumulate --------
__global__ void bn_norm_pool_k(float* __restrict__ a,
                               const float* __restrict__ colsum, const float* __restrict__ colsq,
                               const float* __restrict__ g, const float* __restrict__ be,
                               const int* __restrict__ batch, float* __restrict__ pool) {
  int idx = blockIdx.x * blockDim.x + threadIdx.x;   // N*H threads exactly
  int n = idx >> 9;
  int c = idx & (HH - 1);
  const float invN = 1.0f / (float)NN;
  float mu  = colsum[c] * invN;
  float var = colsq[c] * invN - mu * mu;
  float v = (a[idx] - mu) * rsqrtf(var + 1e-5f) * g[c] + be[c];
  v = fmaxf(v, 0.f);
  a[idx] = v;
  atomicAddF(&pool[(size_t)batch[n] * HH + c], v);
}

// ---------------- graph-size counts ----------------
__global__ void cnt_k(const int* __restrict__ batch, float* __restrict__ cnt) {
  int i = blockIdx.x * blockDim.x + threadIdx.x;
  if (i < NN) atomicAddF(&cnt[batch[i]], 1.0f);
}

// ---------------- head: z=relu((x1+x2)@fcW+fcb); z@fc1W+fc1b; log_softmax ---
__global__ void head_k(const float* __restrict__ pool1, const float* __restrict__ pool2,
                       const float* __restrict__ cnt,
                       const float* __restrict__ fcW, const float* __restrict__ fcb,
                       const float* __restrict__ fc1W, const float* __restrict__ fc1b,
                       float* __restrict__ out) {
  __shared__ float xrow[HH];
  __shared__ float zrow[256];
  __shared__ float logits[10];
  const int r = blockIdx.x;     // 64 blocks
  const int t = threadIdx.x;    // 256 threads
  const float invc = 1.0f / fmaxf(cnt[r], 1.0f);
  xrow[t]       = (pool1[(size_t)r * HH + t]       + pool2[(size_t)r * HH + t])       * invc;
  xrow[t + 256] = (pool1[(size_t)r * HH + t + 256] + pool2[(size_t)r * HH + t + 256]) * invc;
  __syncthreads();
  float acc = fcb[t];
  for (int k = 0; k < HH; ++k) acc += xrow[k] * fcW[k * 256 + t];
  zrow[t] = fmaxf(acc, 0.f);
  __syncthreads();
  if (t < 10) {
    float o = fc1b[t];
    for (int k = 0; k < 256; ++k) o += zrow[k] * fc1W[k * 10 + t];
    logits[t] = o;
  }
  __syncthreads();
  if (t == 0) {
    float mx = logits[0];
    for (int j = 1; j < 10; ++j) mx = fmaxf(mx, logits[j]);
    float se = 0.f;
    for (int j = 0; j < 10; ++j) se += expf(logits[j] - mx);
    float lse = mx + logf(se);
    for (int j = 0; j < 10; ++j) out[r * 10 + j] = logits[j] - lse;
  }
}

// ======================= host-side orchestration =======================
extern "C" void kernel_launch(void* const* d_in, const int* in_sizes, int n_in,
                              void* d_out, int out_size, void* d_ws, size_t ws_size,
                              hipStream_t stream) {
  const float* x   = (const float*)d_in[0];
  const int*   ei  = (const int*)  d_in[1];   // [2, E]
  const int*   bat = (const int*)  d_in[2];
  const float* W1  = (const float*)d_in[3];
  const float* as1 = (const float*)d_in[4];
  const float* ad1 = (const float*)d_in[5];
  const float* b1  = (const float*)d_in[6];
  const float* g1  = (const float*)d_in[7];
  const float* be1 = (const float*)d_in[8];
  const float* W2  = (const float*)d_in[9];
  const float* as2 = (const float*)d_in[10];
  const float* ad2 = (const float*)d_in[11];
  const float* b2  = (const float*)d_in[12];
  const float* g2  = (const float*)d_in[13];
  const float* be2 = (const float*)d_in[14];
  const float* fcW = (const float*)d_in[15];
  const float* fcb = (const float*)d_in[16];
  const float* f1W = (const float*)d_in[17];
  const float* f1b = (const float*)d_in[18];
  float* out = (float*)d_out;

  // workspace carve-up
  float* ws    = (float*)d_ws;
  float* bufA  = ws;                         // N*H f32
  float* bufB  = bufA + (size_t)NN * HH;     // N*H f32
  float* es    = bufB + (size_t)NN * HH;     // N
  float* ed    = es + NN;                    // N
  float* csum  = ed + NN;                    // 512
  float* csq   = csum + HH;                  // 512
  float* pool1 = csq + HH;                   // B*H
  float* pool2 = pool1 + (size_t)BB * HH;    // B*H
  float* cnt   = pool2 + (size_t)BB * HH;    // B
  int* rowptr  = (int*)(cnt + BB);           // N+1
  int* degcur  = rowptr + (NN + 1);          // N (degree, then cursor)
  int* colsrc  = degcur + NN;                // ETOT

  const int TPB = 256;
  const dim3 gEdge((ETOT + TPB - 1) / TPB);
  const dim3 gN((NN + TPB - 1) / TPB);
  const dim3 gNH((NN * HH) / TPB);           // exact: 40000
  const dim3 gGemm(NN / 16);                 // 1250
  const dim3 gNode(NN / 8);                  // 2500

  // ---- graph-size counts ----
  fill_f32_k<<<1, 64, 0, stream>>>(cnt, 0.f, BB);
  cnt_k<<<gN, TPB, 0, stream>>>(bat, cnt);

  // ---- CSR build (shared by both layers) ----
  fill_i32_k<<<gN, TPB, 0, stream>>>(degcur, 0, NN);
  deg_k<<<gEdge, TPB, 0, stream>>>(ei, degcur);
  scan_rowptr_k<<<1, SCAN_T, 0, stream>>>(degcur, rowptr);
  cursor_init_k<<<gN, TPB, 0, stream>>>(rowptr, degcur);
  scatter_k<<<gEdge, TPB, 0, stream>>>(ei, degcur, colsrc);

  // =================== layer 1 ===================
  gemm_wmma_k<<<gGemm, 256, 0, stream>>>(x, W1, bufA, FIN);
  node_scores_k<<<gNode, 256, 0, stream>>>(bufA, as1, ad1, es, ed);
  gat_node_k<<<gNode, 256, 0, stream>>>(rowptr, colsrc, es, ed, bufA, b1, bufB);
  fill_f32_k<<<4, TPB, 0, stream>>>(csum, 0.f, 2 * HH);   // csum+csq contiguous
  bn_stats_k<<<NN / BN_ROWS, HH, 0, stream>>>(bufB, csum, csq);
  fill_f32_k<<<(BB * HH) / TPB, TPB, 0, stream>>>(pool1, 0.f, BB * HH);
  bn_norm_pool_k<<<gNH, TPB, 0, stream>>>(bufB, csum, csq, g1, be1, bat, pool1);

  // =================== layer 2 ===================
  gemm_wmma_k<<<gGemm, 256, 0, stream>>>(bufB, W2, bufA, HH);
  node_scores_k<<<gNode, 256, 0, stream>>>(bufA, as2, ad2, es, ed);
  gat_node_k<<<gNode, 256, 0, stream>>>(rowptr, colsrc, es, ed, bufA, b2, bufB);
  fill_f32_k<<<4, TPB, 0, stream>>>(csum, 0.f, 2 * HH);
  bn_stats_k<<<NN / BN_ROWS, HH, 0, stream>>>(bufB, csum, csq);
  fill_f32_k<<<(BB * HH) / TPB, TPB, 0, stream>>>(pool2, 0.f, BB * HH);
  bn_norm_pool_k<<<gNH, TPB, 0, stream>>>(bufB, csum, csq, g2, be2, bat, pool2);

  // =================== head ===================
  head_k<<<BB, 256, 0, stream>>>(pool1, pool2, cnt, fcW, fcb, f1W, f1b, out);
}